// STPRNN_58995670778382
// MI455X (gfx1250) — compile-verified
//
#include <hip/hip_runtime.h>
#include <cmath>

// STP-RNN forward for MI455X (gfx1250, wave32).
// B=2048, T=1024, D=64, H=16.
// d_out layout (flat, return order): output[B*T] | hidden[B*T*16] | x_cat[B*T*128]
//
// K12 (fused): one 64-thread block per batch b. Threads d=0..63 run the
//   synaptic-depression scan, stream x_cat to global with non-temporal
//   stores, and stage 16x128 row tiles in double-buffered LDS. The two
//   waves alternate computing the 16x128 @ 128x16 projection with 32
//   chained V_WMMA_F32_16X16X4_F32, writing xw into the hidden region of
//   d_out (128 MB -> fits the 192 MB L2; NT hints on streaming traffic
//   keep it resident for K3).
// K3: transposed recurrence h_{t+1}^T = relu(xw^T + Whh^T @ h^T), one wave
//   per 16 batch rows. Whh^T is the constant A operand; the D-layout state
//   converts to next-step B fragments with 8 shfl_xor(16) + selects (no
//   LDS round trip). xw load / hidden store are 2x b128 per lane, in place.
//   The linear readout folds in as a register dot + one cross-half shuffle.

constexpr int B = 2048, T = 1024, D = 64, H = 16;
constexpr long HID_OFF  = (long)B * T;            // after output[B*T]
constexpr long XCAT_OFF = (long)B * T * (1 + H);  // after hidden[B*T*H]

#define TAU_INV (1.0f / 6.0f)
#define UCOEF 0.5f

typedef float v2f __attribute__((ext_vector_type(2)));
typedef float v4f __attribute__((ext_vector_type(4)));
typedef float v8f __attribute__((ext_vector_type(8)));

// LDS tile row stride: 132 floats -> A-fragment ds_load_b64 across 32 lanes
// touches 32 distinct banks (132 mod 64 = 4; halves offset by +2).
constexpr int LDS_STRIDE = 132;

// ---------------------------------------------------------------- kernel 12
__global__ void __launch_bounds__(64)
k12_syn_xcat_xw(const float* __restrict__ inputs,
                const float* __restrict__ syn_x0,
                const float* __restrict__ w_ih,
                const float* __restrict__ bias,
                float* out) {
  __shared__ float tile[2][16 * LDS_STRIDE];  // double-buffered 16x128 tiles

  const int b    = blockIdx.x;
  const int d    = threadIdx.x;      // 0..63
  const int lane = threadIdx.x & 31;
  const int n    = lane & 15;
  const int half = lane >> 4;
  const int wib  = threadIdx.x >> 5; // wave in block: 0 or 1

  // Preload W_ih B-fragments (both waves; each needs them on its turn).
  // Chunk kc covers K rows 4kc..4kc+3: VGPR0 rows k0/k0+2, VGPR1 k0+1/k0+3.
  v2f bf[32];
#pragma unroll
  for (int kc = 0; kc < 32; ++kc) {
    int k0 = 4 * kc + 2 * half;
    bf[kc].x = w_ih[k0 * 16 + n];
    bf[kc].y = w_ih[(k0 + 1) * 16 + n];
  }
  const float bv = bias[n];

  float s = syn_x0[(long)b * D + d];
  const float* ip = inputs + (long)b * T * D + d;
  float* xc = out + XCAT_OFF + (long)b * T * (2 * D) + d;
  float* XW = out + HID_OFF;

  for (int tb = 0; tb < T / 16; ++tb) {
    float* buf = tile[tb & 1];
    // ---- 16 scan steps fill one tile ----
#pragma unroll 4
    for (int tt = 0; tt < 16; ++tt) {
      int t = tb * 16 + tt;
      float x = __builtin_nontemporal_load(ip + (long)t * D);
      float k = TAU_INV + UCOEF * x;
      float sx = s * x;
      __builtin_nontemporal_store(x,  xc + (long)t * (2 * D));
      __builtin_nontemporal_store(sx, xc + (long)t * (2 * D) + D);
      buf[tt * LDS_STRIDE + d]     = x;
      buf[tt * LDS_STRIDE + D + d] = sx;
      s = (TAU_INV - (TAU_INV - s * k) * __expf(-k)) / k;
    }
    __syncthreads();
    // ---- alternating wave computes the 16x16 projection tile ----
    if (wib == (tb & 1)) {
      v8f c = {0.f, 0.f, 0.f, 0.f, 0.f, 0.f, 0.f, 0.f};
      const float* arow = buf + n * LDS_STRIDE;  // A row m = lane&15
#pragma unroll
      for (int kc = 0; kc < 32; ++kc) {
        v2f a = *(const v2f*)(arow + 4 * kc + 2 * half);  // 8B-aligned
        c = __builtin_amdgcn_wmma_f32_16x16x4_f32(
            false, a, false, bf[kc], (short)0, c, false, false);
      }
      long rowbase = (long)b * T + tb * 16;
#pragma unroll
      for (int i = 0; i < 8; ++i)
        XW[(rowbase + i + 8 * half) * 16 + n] = c[i] + bv;  // L2-resident
    }
  }
}

// ---------------------------------------------------------------- kernel 3
__global__ void __launch_bounds__(128)
k3_rnn_wmma(const float* __restrict__ h0,
            const float* __restrict__ w_hh,
            const float* __restrict__ lin_w,
            const float* __restrict__ lin_b,
            float* out) {
  const int lane = threadIdx.x & 31;
  const int n    = lane & 15;
  const int half = lane >> 4;
  const long bbase = ((long)blockIdx.x * 4 + (threadIdx.x >> 5)) * 16;

  // A' = Whh^T fragments (constant): af[kc][c] = Whh[4kc+c+2*half][n]
  v2f af[4];
#pragma unroll
  for (int kc = 0; kc < 4; ++kc) {
    int k0 = 4 * kc + 2 * half;
    af[kc].x = w_hh[k0 * 16 + n];
    af[kc].y = w_hh[(k0 + 1) * 16 + n];
  }
  // Readout weights for this half's hid rows
  v8f lw;
#pragma unroll
  for (int i = 0; i < 8; ++i) lw[i] = lin_w[i + 8 * half];
  const float lb = lin_b[0];

  // Per-lane base pointers: this lane owns batch row b = bbase+n.
  float* rowp = out + HID_OFF + ((long)(bbase + n) * T) * 16 + 8 * half;
  float* op   = out + (long)(bbase + n) * T;

  // State hT in C/D layout: s[i] @ lane L = h[b=bbase+n][hid=i+8*half]
  v8f s;
  {
    const float* hp = h0 + (long)(bbase + n) * H + 8 * half;
    v4f a0 = *(const v4f*)(hp);
    v4f a1 = *(const v4f*)(hp + 4);
#pragma unroll
    for (int i = 0; i < 4; ++i) { s[i] = a0[i]; s[i + 4] = a1[i]; }
  }

  for (int t = 0; t < T; ++t) {
    // C := xw^T tile, two b128 loads per lane (L2-hot)
    v4f c0 = *(const v4f*)(rowp + (long)t * 16);
    v4f c1 = *(const v4f*)(rowp + (long)t * 16 + 4);
    v8f d;
#pragma unroll
    for (int i = 0; i < 4; ++i) { d[i] = c0[i]; d[i + 4] = c1[i]; }

    // B' fragments from state via cross-half shuffles, then accumulate
#pragma unroll
    for (int kc = 0; kc < 4; ++kc) {
      v2f bfrag;
#pragma unroll
      for (int c2 = 0; c2 < 2; ++c2) {
        const int klo = 4 * kc + c2;      // hid needed by lanes 0-15
        const int khi = klo + 2;          // hid needed by lanes 16-31
        float lo = (klo < 8) ? s[klo] : __shfl_xor(s[klo - 8], 16, 32);
        float hi = (khi < 8) ? __shfl_xor(s[khi], 16, 32) : s[khi - 8];
        bfrag[c2] = half ? hi : lo;
      }
      d = __builtin_amdgcn_wmma_f32_16x16x4_f32(
          false, af[kc], false, bfrag, (short)0, d, false, false);
    }

    // ReLU + commit state
#pragma unroll
    for (int i = 0; i < 8; ++i) d[i] = fmaxf(d[i], 0.0f);
    s = d;

    // hidden store (in place over xw), two b128 NT stores
    v4f o0, o1;
#pragma unroll
    for (int i = 0; i < 4; ++i) { o0[i] = d[i]; o1[i] = d[i + 4]; }
    __builtin_nontemporal_store(o0, (v4f*)(rowp + (long)t * 16));
    __builtin_nontemporal_store(o1, (v4f*)(rowp + (long)t * 16 + 4));

    // Fused readout: output[b,t] = hidden . lin_w + lin_b
    float dot = 0.f;
#pragma unroll
    for (int i = 0; i < 8; ++i) dot += d[i] * lw[i];
    dot += __shfl_xor(dot, 16, 32);
    if (half == 0) op[t] = dot + lb;  // wave reconverges before next WMMA
  }
}

// ---------------------------------------------------------------- launcher
extern "C" void kernel_launch(void* const* d_in, const int* in_sizes, int n_in,
                              void* d_out, int out_size, void* d_ws,
                              size_t ws_size, hipStream_t stream) {
  const float* inputs = (const float*)d_in[0];
  const float* syn_x0 = (const float*)d_in[1];
  const float* h0     = (const float*)d_in[2];
  const float* w_ih   = (const float*)d_in[3];
  const float* w_hh   = (const float*)d_in[4];
  const float* bias   = (const float*)d_in[5];
  const float* lin_w  = (const float*)d_in[6];
  const float* lin_b  = (const float*)d_in[7];
  float* out = (float*)d_out;

  // Fused scan + x_cat + input projection: one block per batch row.
  k12_syn_xcat_xw<<<B, 64, 0, stream>>>(inputs, syn_x0, w_ih, bias, out);
  // Recurrence + fused readout: B/16 = 128 waves, 4 waves/block.
  k3_rnn_wmma<<<32, 128, 0, stream>>>(h0, w_hh, lin_w, lin_b, out);
}